// SetConvLayer_9070970929194
// MI455X (gfx1250) — compile-verified
//
#include <hip/hip_runtime.h>
#include <math.h>

#define NPTS 16384
#define FIN  64
#define MCTR 4096
#define KNBR 32
#define RAD2 0.25f
#define D0P  68      // FIN + 3 (rel coords) padded to multiple of 4
#define H1D  64
#define H2D  64
#define H3D  128
#define CANDCAP 96

typedef __attribute__((ext_vector_type(2))) float v2f;
typedef __attribute__((ext_vector_type(8))) float v8f;

// ---------------------------------------------------------------------------
// 1) Farthest point sampling: one workgroup, all positions in LDS (196KB).
//    4095 serial rounds of (distance-update, block argmax).
// ---------------------------------------------------------------------------
__global__ __launch_bounds__(1024) void fps_kernel(const float* __restrict__ pos,
                                                   int* __restrict__ fps_idx) {
  extern __shared__ float sm[];
  float* px = sm;
  float* py = sm + NPTS;
  float* pz = sm + 2 * NPTS;
  float* redV = sm + 3 * NPTS;                 // [1024]
  int*   redI = (int*)(sm + 3 * NPTS + 1024);  // [1024]
  int*   bcast = (int*)(sm + 3 * NPTS + 2048); // [1]

  const int tid = threadIdx.x;
  for (int p = tid; p < NPTS; p += 1024) {
    px[p] = pos[3 * p + 0];
    py[p] = pos[3 * p + 1];
    pz[p] = pos[3 * p + 2];
  }
  float dmin[16];
#pragma unroll
  for (int j = 0; j < 16; ++j) dmin[j] = INFINITY;
  if (tid == 0) fps_idx[0] = 0;
  __syncthreads();

  const int lane = tid & 31, wv = tid >> 5;
  int last = 0;
  for (int t = 1; t < MCTR; ++t) {
    const float cx = px[last], cy = py[last], cz = pz[last];
    float bv = -1.0f;
    int bi = 0x7fffffff;
#pragma unroll
    for (int j = 0; j < 16; ++j) {
      const int p = tid + 1024 * j;
      const float dx = px[p] - cx, dy = py[p] - cy, dz = pz[p] - cz;
      const float d = dx * dx + dy * dy + dz * dz;
      const float dm = fminf(dmin[j], d);
      dmin[j] = dm;
      if (dm > bv || (dm == bv && p < bi)) { bv = dm; bi = p; }
    }
#pragma unroll
    for (int off = 16; off > 0; off >>= 1) {
      const float ov = __shfl_xor(bv, off);
      const int oi = __shfl_xor(bi, off);
      if (ov > bv || (ov == bv && oi < bi)) { bv = ov; bi = oi; }
    }
    if (lane == 0) { redV[wv] = bv; redI[wv] = bi; }
    __syncthreads();
    if (wv == 0) {
      bv = redV[lane];
      bi = redI[lane];
#pragma unroll
      for (int off = 16; off > 0; off >>= 1) {
        const float ov = __shfl_xor(bv, off);
        const int oi = __shfl_xor(bi, off);
        if (ov > bv || (ov == bv && oi < bi)) { bv = ov; bi = oi; }
      }
      if (lane == 0) { bcast[0] = bi; fps_idx[t] = bi; }
    }
    __syncthreads();
    last = bcast[0];
  }
}

// ---------------------------------------------------------------------------
// 2) Ball query: one wave per center. Deterministic candidate collection via
//    ballot + prefix popcount; if >32 in-radius, select 32 smallest d2
//    (tie-break on point index, matching top_k's first-occurrence order).
//    Also emits pos[idx] and batch[idx] output sections.
// ---------------------------------------------------------------------------
__global__ __launch_bounds__(256) void nbr_kernel(const float* __restrict__ pos,
                                                  const int* __restrict__ batch,
                                                  const int* __restrict__ fps_idx,
                                                  int* __restrict__ nbr_idx,
                                                  int* __restrict__ nbr_cnt,
                                                  float* __restrict__ pos_out,
                                                  float* __restrict__ batch_out) {
  __shared__ float cd2[8][CANDCAP];
  __shared__ int cid[8][CANDCAP];
  const int wv = threadIdx.x >> 5, lane = threadIdx.x & 31;
  const int c = blockIdx.x * 8 + wv;
  const int ci = fps_idx[c];
  const float cx = pos[3 * ci], cy = pos[3 * ci + 1], cz = pos[3 * ci + 2];
  if (lane == 0) {
    pos_out[3 * c + 0] = cx;
    pos_out[3 * c + 1] = cy;
    pos_out[3 * c + 2] = cz;
    ((int*)batch_out)[c] = batch[ci];
  }
  int cnt = 0;
  for (int p0 = 0; p0 < NPTS; p0 += 32) {
    const int p = p0 + lane;
    const float dx = pos[3 * p] - cx, dy = pos[3 * p + 1] - cy, dz = pos[3 * p + 2] - cz;
    const float d2 = dx * dx + dy * dy + dz * dz;
    const bool hit = d2 <= RAD2;
    const unsigned mask = (unsigned)__ballot(hit);
    if (hit) {
      const int slot = cnt + __popc(mask & ((1u << lane) - 1u));
      if (slot < CANDCAP) { cd2[wv][slot] = d2; cid[wv][slot] = p; }
    }
    cnt += __popc(mask);
  }
  const int ccl = cnt < CANDCAP ? cnt : CANDCAP;
  if (cnt <= KNBR) {
    if (lane < ccl) nbr_idx[c * KNBR + lane] = cid[wv][lane];
    if (lane == 0) nbr_cnt[c] = ccl;
  } else {
    for (int s = 0; s < KNBR; ++s) {
      float bv = INFINITY;
      int bp = 0x7fffffff, bq = -1;
      for (int q = lane; q < ccl; q += 32) {
        const float v = cd2[wv][q];
        const int pi = cid[wv][q];
        if (v < bv || (v == bv && pi < bp)) { bv = v; bp = pi; bq = q; }
      }
#pragma unroll
      for (int off = 16; off > 0; off >>= 1) {
        const float ov = __shfl_xor(bv, off);
        const int op = __shfl_xor(bp, off);
        const int oq = __shfl_xor(bq, off);
        if (ov < bv || (ov == bv && op < bp)) { bv = ov; bp = op; bq = oq; }
      }
      if (lane == 0) { nbr_idx[c * KNBR + s] = bp; cd2[wv][bq] = INFINITY; }
    }
    if (lane == 0) nbr_cnt[c] = KNBR;
  }
}

// ---------------------------------------------------------------------------
// 3) Per-edge MLP via fp32 WMMA (V_WMMA_F32_16X16X4_F32) + fused masked max.
//    A tile  (16x4 f32): lane = M(0..15) | M(0..15), v2f = K pair {2h,2h+1}
//    B tile  (4x16 f32): lane = N(0..15) | N(0..15), v2f = K pair (mirrored)
//    C/D tile(16x16 f32): v8f, row = 8*(lane>>4)+r, col = lane&15
//    A chunks for both M-tiles are register-resident across all N-tiles;
//    each B chunk is loaded once and feeds both M-tile accumulators, so
//    LDS traffic drops to ~1.2 dwords per WMMA (WMMA-issue bound).
// ---------------------------------------------------------------------------
template <int KDIM, int NCOLS>
__device__ inline void layer_wmma(const float* __restrict__ Hin,
                                  const float* __restrict__ W,
                                  const float* __restrict__ bias,
                                  float* __restrict__ Hout, int lane) {
  constexpr int KCH = KDIM / 4;
  const int lm = lane & 15;
  const int lh = lane >> 4;
  const int koff = 2 * lh;

  // Preload A operand chunks for both 16-row tiles into registers.
  v2f a[2][KCH];
#pragma unroll
  for (int mt = 0; mt < 2; ++mt) {
    const float* ha = Hin + (mt * 16 + lm) * KDIM + koff;
#pragma unroll
    for (int k = 0; k < KCH; ++k) {
      a[mt][k].x = ha[4 * k];
      a[mt][k].y = ha[4 * k + 1];
    }
  }

#pragma unroll
  for (int nt = 0; nt < NCOLS / 16; ++nt) {
    const int bn = nt * 16 + lm;
    v8f acc0 = {};
    v8f acc1 = {};
#pragma unroll
    for (int k = 0; k < KCH; ++k) {
      v2f b;
      const float* wb = W + (4 * k + koff) * NCOLS + bn;
      b.x = wb[0];
      b.y = wb[NCOLS];
      acc0 = __builtin_amdgcn_wmma_f32_16x16x4_f32(false, a[0][k], false, b,
                                                   (short)0, acc0, false, false);
      acc1 = __builtin_amdgcn_wmma_f32_16x16x4_f32(false, a[1][k], false, b,
                                                   (short)0, acc1, false, false);
    }
    const float bval = bias[bn];
#pragma unroll
    for (int r = 0; r < 8; ++r) {
      float v0 = acc0[r] + bval;
      float v1 = acc1[r] + bval;
      v0 = v0 > 0.0f ? v0 : 0.0f;
      v1 = v1 > 0.0f ? v1 : 0.0f;
      Hout[(lh * 8 + r) * NCOLS + bn] = v0;
      Hout[(16 + lh * 8 + r) * NCOLS + bn] = v1;
    }
  }
}

__global__ __launch_bounds__(128) void mlp_kernel(const float* __restrict__ feat,
                                                  const float* __restrict__ pos,
                                                  const float* __restrict__ W1,
                                                  const float* __restrict__ b1,
                                                  const float* __restrict__ W2,
                                                  const float* __restrict__ b2,
                                                  const float* __restrict__ W3,
                                                  const float* __restrict__ b3,
                                                  const int* __restrict__ fps_idx,
                                                  const int* __restrict__ nbr_idx,
                                                  const int* __restrict__ nbr_cnt,
                                                  float* __restrict__ out) {
  extern __shared__ float sm[];
  float* W1p = sm;                    // 68*64 (row 67 zero-padded)
  float* W2p = W1p + D0P * H1D;       // 64*64
  float* W3p = W2p + H1D * H2D;       // 64*128
  float* b1s = W3p + H2D * H3D;       // 64
  float* b2s = b1s + H1D;             // 64
  float* b3s = b2s + H2D;             // 128
  float* hbuf = b3s + H3D;            // 4 waves * (32*68 + 32*128)

  const int tid = threadIdx.x;
  for (int i = tid; i < D0P * H1D; i += 128) W1p[i] = (i < 67 * H1D) ? W1[i] : 0.0f;
  for (int i = tid; i < H1D * H2D; i += 128) W2p[i] = W2[i];
  for (int i = tid; i < H2D * H3D; i += 128) W3p[i] = W3[i];
  if (tid < H1D) b1s[tid] = b1[tid];
  if (tid < H2D) b2s[tid] = b2[tid];
  if (tid < H3D) b3s[tid] = b3[tid];
  __syncthreads();

  const int wv = tid >> 5, lane = tid & 31;
  const int c = blockIdx.x * 4 + wv;
  float* bufA = hbuf + wv * (32 * D0P + 32 * H3D);
  float* bufB = bufA + 32 * D0P;

  const int cnt = nbr_cnt[c];
  const int ci = fps_idx[c];
  const float cx = pos[3 * ci], cy = pos[3 * ci + 1], cz = pos[3 * ci + 2];

  // Gather H0 = [x_j || (pos_j - center) || 0-pad]   (lane = neighbor row)
  {
    float* row = bufA + lane * D0P;
    if (lane < cnt) {
      const int n = nbr_idx[c * KNBR + lane];
      const float* fr = feat + n * FIN;
      for (int f = 0; f < FIN; ++f) row[f] = fr[f];
      row[64] = pos[3 * n + 0] - cx;
      row[65] = pos[3 * n + 1] - cy;
      row[66] = pos[3 * n + 2] - cz;
      row[67] = 0.0f;
    } else {
      for (int f = 0; f < D0P; ++f) row[f] = 0.0f;
    }
  }
  __syncthreads();
  layer_wmma<D0P, H1D>(bufA, W1p, b1s, bufB, lane);   // 32x68 @ 68x64
  __syncthreads();
  layer_wmma<H1D, H2D>(bufB, W2p, b2s, bufA, lane);   // 32x64 @ 64x64
  __syncthreads();
  layer_wmma<H2D, H3D>(bufA, W3p, b3s, bufB, lane);   // 32x64 @ 64x128
  __syncthreads();

  // Masked max over valid neighbor rows (>=1 valid: center is its own nbr)
  const int vrows = cnt < KNBR ? cnt : KNBR;
  for (int cg = 0; cg < 4; ++cg) {
    const int col = lane + 32 * cg;
    float m = -INFINITY;
    for (int j = 0; j < vrows; ++j) m = fmaxf(m, bufB[j * H3D + col]);
    out[c * H3D + col] = m;
  }
}

// ---------------------------------------------------------------------------
extern "C" void kernel_launch(void* const* d_in, const int* in_sizes, int n_in,
                              void* d_out, int out_size, void* d_ws, size_t ws_size,
                              hipStream_t stream) {
  const float* feat = (const float*)d_in[0];
  const float* pos = (const float*)d_in[1];
  const int* batch = (const int*)d_in[2];
  const float* W1 = (const float*)d_in[3];
  const float* b1 = (const float*)d_in[4];
  const float* W2 = (const float*)d_in[5];
  const float* b2 = (const float*)d_in[6];
  const float* W3 = (const float*)d_in[7];
  const float* b3 = (const float*)d_in[8];

  float* out = (float*)d_out;            // [4096*128]
  float* pos_out = out + MCTR * H3D;     // [4096*3]
  float* batch_out = pos_out + MCTR * 3; // [4096] (int bits)

  int* fps_idx = (int*)d_ws;             // [4096]
  int* nbr_cnt = fps_idx + MCTR;         // [4096]
  int* nbr_idx = nbr_cnt + MCTR;         // [4096*32]

  const size_t fps_lds = (size_t)(3 * NPTS + 2048 + 64) * sizeof(float);   // ~205 KB
  const size_t mlp_lds = (size_t)(D0P * H1D + H1D * H2D + H2D * H3D +
                                  H1D + H2D + H3D +
                                  4 * (32 * D0P + 32 * H3D)) * sizeof(float); // ~164 KB
  hipFuncSetAttribute((const void*)fps_kernel,
                      hipFuncAttributeMaxDynamicSharedMemorySize, (int)fps_lds);
  hipFuncSetAttribute((const void*)mlp_kernel,
                      hipFuncAttributeMaxDynamicSharedMemorySize, (int)mlp_lds);

  fps_kernel<<<1, 1024, fps_lds, stream>>>(pos, fps_idx);
  nbr_kernel<<<MCTR / 8, 256, 0, stream>>>(pos, batch, fps_idx, nbr_idx, nbr_cnt,
                                           pos_out, batch_out);
  mlp_kernel<<<MCTR / 4, 128, mlp_lds, stream>>>(feat, pos, W1, b1, W2, b2, W3, b3,
                                                 fps_idx, nbr_idx, nbr_cnt, out);
}